// SpatialHyperedgeMP_59227599012106
// MI455X (gfx1250) — compile-verified
//
#include <hip/hip_runtime.h>
#include <math.h>

#define NN 8192   // nodes / hyperedges
#define DD 512    // feature dim
#define KC 64     // K-chunk staged in LDS
#define LDS_STRIDE 66  // 64 + 2 pad: keeps 8B alignment and spreads banks

typedef __attribute__((ext_vector_type(2))) float v2f;
typedef __attribute__((ext_vector_type(8))) float v8f;

// ---------------------------------------------------------------------------
// Phase 1: per-row stats of incidence_m.
//   cnt_i  = #(x>0) in row i
//   invn_i = 1 / max(sqrt(cnt_i), 1e-12)           (head row L2 normalizer)
//   deg_i  = rowsum(inc)_i + cnt_i * invn_i        (edge degree)
// One 256-thread block per row, float4 loads, LDS tree reduction.
// ---------------------------------------------------------------------------
__global__ __launch_bounds__(256) void row_stats_kernel(
    const float* __restrict__ inc,
    float* __restrict__ invnorm,
    float* __restrict__ invdeg) {
  const int row = blockIdx.x;
  const int tid = threadIdx.x;
  const float4* p4 = (const float4*)(inc + (size_t)row * NN);
  float sum = 0.f, cnt = 0.f;
#pragma unroll
  for (int j = tid; j < NN / 4; j += 256) {
    float4 v = p4[j];
    sum += (v.x + v.y) + (v.z + v.w);
    cnt += (v.x > 0.f ? 1.f : 0.f) + (v.y > 0.f ? 1.f : 0.f) +
           (v.z > 0.f ? 1.f : 0.f) + (v.w > 0.f ? 1.f : 0.f);
  }
  __shared__ float ssum[256];
  __shared__ float scnt[256];
  ssum[tid] = sum;
  scnt[tid] = cnt;
  __syncthreads();
  for (int s = 128; s > 0; s >>= 1) {
    if (tid < s) {
      ssum[tid] += ssum[tid + s];
      scnt[tid] += scnt[tid + s];
    }
    __syncthreads();
  }
  if (tid == 0) {
    const float c = scnt[0];
    const float s = ssum[0];
    const float invn = 1.f / fmaxf(sqrtf(c), 1e-12f);
    const float deg = s + c * invn;
    invnorm[row] = invn;
    invdeg[row] = 1.f / deg;
  }
}

// ---------------------------------------------------------------------------
// Phase 2: out = (A @ cur) * invdeg,  A[i][j] = inc[i][j] + (inc>0)*invn_i
//
// Block = 256 threads (8 wave32), owns 32 output rows x all 512 cols.
// Wave w owns cols [w*64, w*64+64): acc = 2(M-tiles) x 4(N-tiles) of v8f.
// K loop: stage transformed 32xKC A-chunk into padded LDS once per block,
// then 16 k-steps of V_WMMA_F32_16X16X4_F32 per chunk.
//
// Fragment layouts (ISA 7.12.2), lane = 16*h + l (h=half, l=0..15):
//   A 16x4 : VGPR0 = K(2h), VGPR1 = K(2h+1), M = l   -> ds_load_b64 per tile
//   B 4x16 : VGPR0 = K(2h), VGPR1 = K(2h+1), N = l   -> 2 global b32 per tile
//   C 16x16: VGPR r -> M = r + 8h, N = l
// ---------------------------------------------------------------------------
__global__ __launch_bounds__(256) void hyperedge_gemm_kernel(
    const float* __restrict__ cur,
    const float* __restrict__ inc,
    const float* __restrict__ invnorm,
    const float* __restrict__ invdeg,
    float* __restrict__ out) {
  __shared__ float As[32 * LDS_STRIDE];

  const int tid = threadIdx.x;
  const int wave = tid >> 5;
  const int lane = tid & 31;
  const int l = lane & 15;
  const int h = lane >> 4;

  const int rowBase = blockIdx.x * 32;
  const int colBase = wave * 64;

  // Cooperative staging coords: thread -> (row, 8-float column group).
  const int srow = tid >> 3;        // 0..31
  const int scol0 = (tid & 7) * 8;  // 0,8,...,56
  const float sinv = invnorm[rowBase + srow];
  const float* srcRow = inc + (size_t)(rowBase + srow) * NN + scol0;
  float* dstRow = As + srow * LDS_STRIDE + scol0;  // 8B-aligned (stride even)

  v8f acc[2][4] = {};

  for (int kc = 0; kc < NN; kc += KC) {
    __syncthreads();  // previous chunk's ds reads done before overwrite

    // --- stage 32 x KC A-chunk with the head transform applied once ---
    const float4* s4 = (const float4*)(srcRow + kc);
    float4 x0 = s4[0];
    float4 x1 = s4[1];
    v2f t01, t23, t45, t67;
    t01.x = x0.x > 0.f ? x0.x + sinv : x0.x;
    t01.y = x0.y > 0.f ? x0.y + sinv : x0.y;
    t23.x = x0.z > 0.f ? x0.z + sinv : x0.z;
    t23.y = x0.w > 0.f ? x0.w + sinv : x0.w;
    t45.x = x1.x > 0.f ? x1.x + sinv : x1.x;
    t45.y = x1.y > 0.f ? x1.y + sinv : x1.y;
    t67.x = x1.z > 0.f ? x1.z + sinv : x1.z;
    t67.y = x1.w > 0.f ? x1.w + sinv : x1.w;
    v2f* d2 = (v2f*)dstRow;
    d2[0] = t01;
    d2[1] = t23;
    d2[2] = t45;
    d2[3] = t67;
    __syncthreads();

    // warm L2/WGP$ for the next incidence chunk (global_prefetch_b8)
    if (kc + KC < NN) {
      __builtin_prefetch(srcRow + kc + KC, 0, 0);
    }

    // --- compute: 16 k-steps of 4 over this chunk ---
#pragma unroll
    for (int ko = 0; ko < KC; ko += 4) {
      v2f a[2];
#pragma unroll
      for (int mi = 0; mi < 2; ++mi) {
        a[mi] = *(const v2f*)(As + (16 * mi + l) * LDS_STRIDE + ko + 2 * h);
      }
      const int kg = kc + ko + 2 * h;  // this half-wave's K pair base
      const float* b0 = cur + (size_t)kg * DD + colBase + l;
#pragma unroll
      for (int t = 0; t < 4; ++t) {
        v2f b;
        b.x = b0[t * 16];       // row kg
        b.y = b0[t * 16 + DD];  // row kg+1
#pragma unroll
        for (int mi = 0; mi < 2; ++mi) {
          acc[mi][t] = __builtin_amdgcn_wmma_f32_16x16x4_f32(
              false, a[mi], false, b, (short)0, acc[mi][t], false, false);
        }
      }
    }
  }

  // --- epilogue: scale by 1/degree and store ---
#pragma unroll
  for (int mi = 0; mi < 2; ++mi) {
#pragma unroll
    for (int r = 0; r < 8; ++r) {
      const int row = rowBase + 16 * mi + 8 * h + r;
      const float idg = invdeg[row];  // broadcast load (uniform per half)
      float* orow = out + (size_t)row * DD + colBase + l;
#pragma unroll
      for (int t = 0; t < 4; ++t) {
        orow[t * 16] = acc[mi][t][r] * idg;
      }
    }
  }
}

extern "C" void kernel_launch(void* const* d_in, const int* in_sizes, int n_in,
                              void* d_out, int out_size, void* d_ws,
                              size_t ws_size, hipStream_t stream) {
  const float* cur = (const float*)d_in[0];        // [N, D] fp32
  const float* inc = (const float*)d_in[1];        // [N, N] fp32
  float* out = (float*)d_out;                      // [N, D] fp32
  float* invnorm = (float*)d_ws;                   // [N]
  float* invdeg = invnorm + NN;                    // [N]

  row_stats_kernel<<<NN, 256, 0, stream>>>(inc, invnorm, invdeg);
  hyperedge_gemm_kernel<<<NN / 32, 256, 0, stream>>>(cur, inc, invnorm, invdeg,
                                                     out);
}